// CnnSnn_69355131895959
// MI455X (gfx1250) — compile-verified
//
#include <hip/hip_runtime.h>
#include <math.h>

typedef float v2f __attribute__((ext_vector_type(2)));
typedef float v8f __attribute__((ext_vector_type(8)));

#define BATCH   1024
#define C_IN    64
#define C_OUT   512
#define WS_LEN  512
#define KW      32
#define STRIDE_ 16
#define T_STEPS 31
#define KTOT    (C_IN * KW)   // 2048
#define BN_EPS  1e-5f
#define TILE_B  16
#define LSTR    516           // 512 + 4 pad: lane m -> bank 4m (+kk), conflict-free b64 LDS reads

__device__ __forceinline__ v8f wmma_f32(v2f a, v2f b, v8f c) {
    // D = A(16x4 f32) * B(4x16 f32) + C(16x16 f32)
    return __builtin_amdgcn_wmma_f32_16x16x4_f32(false, a, false, b, (short)0, c, false, false);
}

// -------- Kernel A: Conv1d + BatchNorm as implicit GEMM --------------------
// out[t, b, c] = BN( sum_{ci,k} x[b,ci,t*16+k] * w[c,ci,k] + cb[c] )
// Grid: (1024 batches, 2 channel-groups of 256). Block: 256 = 8 waves.
// Wave -> 16(t) x 64(c): 4 accumulators sharing each A-fragment load
// (1 A-load + 4 B-loads per 4 WMMAs instead of 8 loads per 4).
__global__ __launch_bounds__(256) void conv_bn_kernel(
    const float* __restrict__ x,  const float* __restrict__ w,
    const float* __restrict__ cb, const float* __restrict__ g,
    const float* __restrict__ bb_, const float* __restrict__ mu,
    const float* __restrict__ var, float* __restrict__ seq)
{
    const int b     = blockIdx.x;
    const int wave  = threadIdx.x >> 5;
    const int lane  = threadIdx.x & 31;
    const int rt    = wave & 1;              // t rows 0-15 / 16-31
    const int cg    = wave >> 1;             // 0..3 -> 64 channels each
    const int m     = lane & 15;             // A-row (t) / B-col (c)
    const int kh    = lane >> 4;             // K half: lanes>=16 hold K+2,K+3
    const int cbase = blockIdx.y * 256 + cg * 64;

    int tA = rt * 16 + m;
    if (tA > T_STEPS - 1) tA = T_STEPS - 1;  // clamp pad row (result discarded)
    const float* xb = x + (size_t)b * (C_IN * WS_LEN) + tA * STRIDE_;
    const float* wc0 = w + (size_t)(cbase +  0 + m) * KTOT;
    const float* wc1 = w + (size_t)(cbase + 16 + m) * KTOT;
    const float* wc2 = w + (size_t)(cbase + 32 + m) * KTOT;
    const float* wc3 = w + (size_t)(cbase + 48 + m) * KTOT;

    v8f acc0 = {}, acc1 = {}, acc2 = {}, acc3 = {};
    #pragma unroll 4
    for (int k0 = 0; k0 < KTOT; k0 += 4) {
        int kk = k0 + 2 * kh;                // kk even -> 8B aligned float2
        int ci = kk >> 5;
        int k  = kk & 31;
        v2f a  = *(const v2f*)(xb + ci * WS_LEN + k);   // A[t][kk..kk+1], shared x4
        v2f b0 = *(const v2f*)(wc0 + kk);
        v2f b1 = *(const v2f*)(wc1 + kk);
        v2f b2 = *(const v2f*)(wc2 + kk);
        v2f b3 = *(const v2f*)(wc3 + kk);
        acc0 = wmma_f32(a, b0, acc0);
        acc1 = wmma_f32(a, b1, acc1);
        acc2 = wmma_f32(a, b2, acc2);
        acc3 = wmma_f32(a, b3, acc3);
    }

    v8f accs[4] = {acc0, acc1, acc2, acc3};
    #pragma unroll
    for (int j = 0; j < 4; ++j) {
        const int c = cbase + j * 16 + m;
        const float scale = g[c] * rsqrtf(var[c] + BN_EPS);
        const float shift = bb_[c] - mu[c] * scale;
        const float bias  = cb[c];
        #pragma unroll
        for (int i = 0; i < 8; ++i) {
            int tt = rt * 16 + i + 8 * kh;   // D row -> time step
            if (tt < T_STEPS) {
                float v = (accs[j][i] + bias) * scale + shift;
                seq[(size_t)tt * (BATCH * C_OUT) + (size_t)b * C_OUT + c] = v;
            }
        }
    }
}

// -------- Kernel B: 31-step SNN scan -------------------------------------
// 64 workgroups x 16 batch rows. Per step: encoder leaky (regs, spikes->LDS),
// FC1 via fp32 WMMA (4 col-tiles/wave share each LDS A-load; hidden membrane
// lives in the WMMA accumulator lanes), FC2 parallel across all 8 waves with
// butterfly shuffle reduction. Dynamic LDS: spkE + spkH = 2*16*516 floats.
__global__ __launch_bounds__(256) void snn_scan_kernel(
    const float* __restrict__ seq,
    const float* __restrict__ fc1w, const float* __restrict__ fc1b,
    const float* __restrict__ fc2w, const float* __restrict__ fc2b,
    const float* __restrict__ betaE, const float* __restrict__ betaH,
    const float* __restrict__ betaO, float* __restrict__ out)
{
    extern __shared__ float lds[];
    float* spkE = lds;                       // [16][LSTR]
    float* spkH = lds + TILE_B * LSTR;       // [16][LSTR]

    const int tid  = threadIdx.x;
    const int wave = tid >> 5;
    const int lane = tid & 31;
    const int m    = lane & 15;
    const int kh   = lane >> 4;
    const int b0   = blockIdx.x * TILE_B;

    float em[32];                            // encoder membrane: (b,c) = flat j*256+tid
    #pragma unroll
    for (int j = 0; j < 32; ++j) em[j] = 0.f;
    float hm[4][8];                          // hidden membrane in WMMA-D lane layout
    #pragma unroll
    for (int tl = 0; tl < 4; ++tl)
        #pragma unroll
        for (int i = 0; i < 8; ++i) hm[tl][i] = 0.f;
    float om[4];                             // output membrane, pair = wave*4+p (replicated per lane)
    #pragma unroll
    for (int p = 0; p < 4; ++p) om[p] = 0.f;

    // per-wave FC1 column layout: wave covers channels [wave*64, wave*64+64)
    const float* wr0 = fc1w + (size_t)(wave * 64 +  0 + m) * C_OUT;
    const float* wr1 = fc1w + (size_t)(wave * 64 + 16 + m) * C_OUT;
    const float* wr2 = fc1w + (size_t)(wave * 64 + 32 + m) * C_OUT;
    const float* wr3 = fc1w + (size_t)(wave * 64 + 48 + m) * C_OUT;
    const float* ar  = spkE + m * LSTR;      // A-row = batch row m

    float* outOm  = out;                               // [1024,2]
    float* outSpk = out + BATCH * 2;                   // [31,1024,2]
    float* outMem = out + BATCH * 2 + T_STEPS * BATCH * 2;

    for (int t = 0; t < T_STEPS; ++t) {
        // ---- encoder neurons (element-wise, coalesced) ----
        const float* xt = seq + (size_t)t * (BATCH * C_OUT) + (size_t)b0 * C_OUT;
        #pragma unroll
        for (int j = 0; j < 32; ++j) {
            int flat = j * 256 + tid;
            int bb = flat >> 9;
            int c  = flat & 511;
            float beta = fminf(fmaxf(betaE[c], 0.f), 1.f);
            float prev = em[j];
            float reset = prev > 1.f ? 1.f : 0.f;      // reset from PREVIOUS mem
            float mn = fmaf(beta, prev, xt[flat]) - reset;
            em[j] = mn;
            spkE[bb * LSTR + c] = mn > 1.f ? 1.f : 0.f;
        }
        __syncthreads();

        // ---- FC1 GEMM (spikes @ fc1_w^T): 4 col-tiles share each A load ----
        v8f acc0 = {}, acc1 = {}, acc2 = {}, acc3 = {};
        #pragma unroll 4
        for (int k0 = 0; k0 < C_OUT; k0 += 4) {
            int kk = k0 + 2 * kh;
            v2f a  = *(const v2f*)(ar + kk);           // LDS, shared x4
            v2f b0v = *(const v2f*)(wr0 + kk);
            v2f b1v = *(const v2f*)(wr1 + kk);
            v2f b2v = *(const v2f*)(wr2 + kk);
            v2f b3v = *(const v2f*)(wr3 + kk);
            acc0 = wmma_f32(a, b0v, acc0);
            acc1 = wmma_f32(a, b1v, acc1);
            acc2 = wmma_f32(a, b2v, acc2);
            acc3 = wmma_f32(a, b3v, acc3);
        }
        v8f accs[4] = {acc0, acc1, acc2, acc3};

        // ---- hidden neurons (state in WMMA accumulator lanes) ----
        #pragma unroll
        for (int tl = 0; tl < 4; ++tl) {
            const int c = wave * 64 + tl * 16 + m;
            float beta = fminf(fmaxf(betaH[c], 0.f), 1.f);
            float bias = fc1b[c];
            #pragma unroll
            for (int i = 0; i < 8; ++i) {
                int br = i + 8 * kh;                   // batch row of D element
                float prev = hm[tl][i];
                float reset = prev > 1.f ? 1.f : 0.f;
                float mn = fmaf(beta, prev, accs[tl][i] + bias) - reset;
                hm[tl][i] = mn;
                spkH[br * LSTR + c] = mn > 1.f ? 1.f : 0.f;
            }
        }
        __syncthreads();

        // ---- FC2 (512 -> 2) + output neurons: 32 (b,o) pairs over 8 waves ----
        #pragma unroll
        for (int p = 0; p < 4; ++p) {
            const int pair = wave * 4 + p;             // 0..31
            const int bb = pair >> 1, o = pair & 1;
            const float* sh = spkH + bb * LSTR;
            const float* w2 = fc2w + o * C_OUT;
            float s = 0.f;
            #pragma unroll
            for (int k = 0; k < 16; ++k) {
                int idx = k * 32 + lane;
                s = fmaf(sh[idx], w2[idx], s);
            }
            #pragma unroll
            for (int off = 16; off > 0; off >>= 1)     // butterfly: every lane gets total
                s += __shfl_xor(s, off, 32);
            float beta = fminf(fmaxf(betaO[o], 0.f), 1.f);
            float prev = om[p];
            float reset = prev > 1.f ? 1.f : 0.f;
            float mn = fmaf(beta, prev, s + fc2b[o]) - reset;
            om[p] = mn;                                // uniform across lanes
            if (lane == 0) {
                size_t oi = (size_t)t * (BATCH * 2) + (size_t)(b0 + bb) * 2 + o;
                outSpk[oi] = mn > 1.f ? 1.f : 0.f;
                outMem[oi] = mn;
                if (t == T_STEPS - 1) outOm[(b0 + bb) * 2 + o] = mn;
            }
        }
        __syncthreads();
    }
}

extern "C" void kernel_launch(void* const* d_in, const int* in_sizes, int n_in,
                              void* d_out, int out_size, void* d_ws, size_t ws_size,
                              hipStream_t stream) {
    const float* x     = (const float*)d_in[0];
    const float* convw = (const float*)d_in[1];
    const float* convb = (const float*)d_in[2];
    const float* bng   = (const float*)d_in[3];
    const float* bnb   = (const float*)d_in[4];
    const float* bnm   = (const float*)d_in[5];
    const float* bnv   = (const float*)d_in[6];
    const float* fc1w  = (const float*)d_in[7];
    const float* fc1b  = (const float*)d_in[8];
    const float* fc2w  = (const float*)d_in[9];
    const float* fc2b  = (const float*)d_in[10];
    const float* betaE = (const float*)d_in[11];
    const float* betaH = (const float*)d_in[12];
    const float* betaO = (const float*)d_in[13];

    float* seq = (float*)d_ws;               // [31][1024][512] f32 = 62 MB (fits L2)
    float* out = (float*)d_out;

    dim3 gridA(BATCH, 2);                    // (1024 batches, 2 groups of 256 channels)
    conv_bn_kernel<<<gridA, 256, 0, stream>>>(x, convw, convb, bng, bnb, bnm, bnv, seq);

    size_t shmem = (size_t)2 * TILE_B * LSTR * sizeof(float);   // 66,048 B
    snn_scan_kernel<<<BATCH / TILE_B, 256, shmem, stream>>>(
        seq, fc1w, fc1b, fc2w, fc2b, betaE, betaH, betaO, out);
}